// STGCNLayer_67216238183256
// MI455X (gfx1250) — compile-verified
//
#include <hip/hip_runtime.h>
#include <cstdint>

typedef __attribute__((ext_vector_type(2))) float v2f;
typedef __attribute__((ext_vector_type(8))) float v8f;

#define WMMA4(a, b, c) \
  __builtin_amdgcn_wmma_f32_16x16x4_f32(false, (a), false, (b), (short)0, (c), false, false)

// Problem dims (reference: N,C,T,V,P,O = 32,64,300,25,3,128, stride 2)
constexpr int Nb   = 32;
constexpr int Cc   = 64;
constexpr int Tt   = 300;
constexpr int Vv   = 25;
constexpr int Pp   = 3;
constexpr int Oo   = 128;
constexpr int TOUT = 150;            // (300 + 8 - 9)/2 + 1
constexpr int TV   = Tt * Vv;        // 7500
constexpr int COLS = TOUT * Vv;      // 3750
constexpr int KT   = Oo * 9;         // 1152 (conv reduction dim)
constexpr float EPSF = 1e-5f;

// Column-tile counts
constexpr int CT_GCN = (TV   + 15) / 16;  // 469
constexpr int CT_TCN = (COLS + 15) / 16;  // 235

// Workspace layout (float offsets)
constexpr size_t ASUM_OFF = 0;                                    // 28*32 padded A_sum
constexpr size_t MEAN_OFF = 1024;                                 // 128
constexpr size_t RSTD_OFF = 1152;                                 // 128
constexpr size_t AGG_OFF  = 2048;                                 // N*C*T*V (reused for x_tcn)
constexpr size_t XGCN_OFF = AGG_OFF + (size_t)Nb * Cc * Tt * Vv;  // N*TV*O (channel-innermost)
constexpr size_t TWP_OFF  = XGCN_OFF + (size_t)Nb * TV * Oo;      // O*KT repacked tcn weights

// ---------------------------------------------------------------------------
// Kernel R: repack tcn_w from (o, i*9+k) to tap-major (o, k*128+i) so the
// conv K-loop uses shift/and index math and contiguous A-fragment loads.
// ---------------------------------------------------------------------------
__global__ void k_repack(const float* __restrict__ tw, float* __restrict__ twp) {
  const int idx = blockIdx.x * blockDim.x + threadIdx.x;
  if (idx >= Oo * KT) return;
  const int o = idx / KT, rest = idx % KT;   // rest = k*128 + i
  const int k = rest >> 7;
  const int i = rest & 127;
  twp[idx] = tw[o * KT + i * 9 + k];
}

// ---------------------------------------------------------------------------
// Kernel 0: A_sum[v][w] = sum_p A_parts[p][v][w] * edge_importance[p],
// zero-padded to 28 x 32.
// ---------------------------------------------------------------------------
__global__ void k_asum(const float* __restrict__ A, const float* __restrict__ ei,
                       float* __restrict__ asum_pad) {
  for (int idx = threadIdx.x; idx < 28 * 32; idx += blockDim.x) {
    int v = idx >> 5, w = idx & 31;
    float s = 0.f;
    if (v < Vv && w < Vv) {
      #pragma unroll
      for (int p = 0; p < Pp; ++p) s += A[(p * Vv + v) * Vv + w] * ei[p];
    }
    asum_pad[idx] = s;
  }
}

// ---------------------------------------------------------------------------
// Kernel 1: agg[(n,c,t), w] = sum_v x[(n,c,t), v] * A_sum[v][w]
// GEMM: M = N*C*T rows, K = 25 (padded 28), Ncols = 25 (2 col tiles).
// A_sum staged in LDS; one 16x16 tile per wave via V_WMMA_F32_16X16X4_F32.
// ---------------------------------------------------------------------------
__global__ __launch_bounds__(256) void k_agg(const float* __restrict__ x,
                                             const float* __restrict__ asum_pad,
                                             float* __restrict__ agg) {
  __shared__ float asum[28 * 32];
  for (int i = threadIdx.x; i < 28 * 32; i += blockDim.x) asum[i] = asum_pad[i];
  __syncthreads();

  const int L  = threadIdx.x & 31;
  const int wv = (blockIdx.x * blockDim.x + threadIdx.x) >> 5;
  const int colTile = wv & 1;
  const int rowTile = wv >> 1;

  const int m   = rowTile * 16 + (L & 15);   // global row (n,c,t)
  const int kh  = (L >> 4) * 2;              // lane-half K offset
  const int col = colTile * 16 + (L & 15);   // output column w

  v8f acc = {};
  #pragma unroll
  for (int s = 0; s < 7; ++s) {              // K = 28 / 4
    const int k0 = s * 4 + kh;
    v2f a, b;
    a.x = (k0     < Vv) ? x[(size_t)m * Vv + k0]     : 0.f;
    a.y = (k0 + 1 < Vv) ? x[(size_t)m * Vv + k0 + 1] : 0.f;
    b.x = asum[(k0    ) * 32 + col];
    b.y = asum[(k0 + 1) * 32 + col];
    acc = WMMA4(a, b, acc);
  }

  if (col < Vv) {
    const int rbase = rowTile * 16 + 8 * (L >> 4);
    #pragma unroll
    for (int r = 0; r < 8; ++r)
      agg[(size_t)(rbase + r) * Vv + col] = acc[r];
  }
}

// ---------------------------------------------------------------------------
// Kernel 2: x_gcn[n,tv,o] = sum_c gcn_w[o,c] * agg[n,c,tv] + gcn_b[o]
// Per-n GEMM (128 x 64) x (64 x 7500). K = 64 exact.
// NOTE: output is channel-INNERMOST so the conv's B-fragments load b64.
// Epilogue per lane writes 8 consecutive floats (merged wide stores).
// ---------------------------------------------------------------------------
__global__ __launch_bounds__(256) void k_gcn(const float* __restrict__ agg,
                                             const float* __restrict__ gw,
                                             const float* __restrict__ gb,
                                             float* __restrict__ xg) {
  const int L  = threadIdx.x & 31;
  const int wv = (blockIdx.x * blockDim.x + threadIdx.x) >> 5;
  const int colTile = wv % CT_GCN;
  const int rem     = wv / CT_GCN;
  const int oTile   = rem & 7;
  const int n       = rem >> 3;
  if (n >= Nb) return;

  const int o_a  = oTile * 16 + (L & 15);
  const int col  = colTile * 16 + (L & 15);
  const bool ok  = col < TV;
  const int kh   = (L >> 4) * 2;
  const float* Bn = agg + (size_t)n * Cc * TV;

  v8f acc = {};
  #pragma unroll 4
  for (int s = 0; s < Cc / 4; ++s) {
    const int k0 = s * 4 + kh;                            // even -> 8B aligned
    const v2f a = *(const v2f*)(gw + o_a * Cc + k0);
    v2f b;
    b.x = ok ? Bn[(size_t)(k0    ) * TV + col] : 0.f;
    b.y = ok ? Bn[(size_t)(k0 + 1) * TV + col] : 0.f;
    acc = WMMA4(a, b, acc);
  }

  if (ok) {
    const int obase = oTile * 16 + 8 * (L >> 4);
    float* dst = xg + ((size_t)n * TV + col) * Oo + obase;
    #pragma unroll
    for (int r = 0; r < 8; ++r) dst[r] = acc[r] + gb[obase + r];
  }
}

// ---------------------------------------------------------------------------
// Kernel 3: temporal conv (9 taps, stride 2, pad 4) as implicit-im2col GEMM.
// Tap-major K order: kappa = k*128 + i. Every 4-wide K-step stays inside one
// tap, so both lane-half K-values share t = 2t'+k-4 and load one b64 from the
// channel-innermost x_gcn. Weights come repacked (contiguous b64).
// x_tcn layout: [n][o][t',v] (for BN reduction + epilogue).
// ---------------------------------------------------------------------------
__global__ __launch_bounds__(256) void k_tcn(const float* __restrict__ xg,
                                             const float* __restrict__ twp,
                                             const float* __restrict__ tb,
                                             float* __restrict__ xt) {
  const int L  = threadIdx.x & 31;
  const int wv = (blockIdx.x * blockDim.x + threadIdx.x) >> 5;
  const int colTile = wv % CT_TCN;
  const int rem     = wv / CT_TCN;
  const int oTile   = rem & 7;
  const int n       = rem >> 3;
  if (n >= Nb) return;

  const int o_a = oTile * 16 + (L & 15);
  const int col = colTile * 16 + (L & 15);
  const bool ok = col < COLS;
  const int tp  = col / Vv;
  const int v   = col % Vv;
  const int kh  = (L >> 4) * 2;
  const float* Bn = xg + (size_t)n * TV * Oo;

  v8f acc = {};
  #pragma unroll 4
  for (int s = 0; s < KT / 4; ++s) {         // 288 k-steps
    const int k0 = s * 4 + kh;               // even
    const v2f a = *(const v2f*)(twp + (size_t)o_a * KT + k0);
    const int tap = k0 >> 7;                 // 0..8
    const int i0  = k0 & 127;                // even -> 8B aligned
    const int t   = 2 * tp + tap - 4;
    v2f b = {0.f, 0.f};
    if (ok && (unsigned)t < (unsigned)Tt)
      b = *(const v2f*)(Bn + ((size_t)t * Vv + v) * Oo + i0);
    acc = WMMA4(a, b, acc);
  }

  if (ok) {
    #pragma unroll
    for (int r = 0; r < 8; ++r) {
      const int o = oTile * 16 + r + 8 * (L >> 4);
      xt[((size_t)n * Oo + o) * COLS + col] = acc[r] + tb[o];
    }
  }
}

// ---------------------------------------------------------------------------
// Kernel 4: BN statistics per channel o over (n, t', v): mean + rsqrt(var+eps).
// ---------------------------------------------------------------------------
__global__ __launch_bounds__(256) void k_bnstats(const float* __restrict__ xt,
                                                 float* __restrict__ mean,
                                                 float* __restrict__ rstd) {
  const int o = blockIdx.x;
  constexpr int PER = Nb * COLS;  // 120000
  float s = 0.f, sq = 0.f;
  for (int i = threadIdx.x; i < PER; i += blockDim.x) {
    const int n = i / COLS, c = i % COLS;
    const float val = xt[((size_t)n * Oo + o) * COLS + c];
    s += val;
    sq += val * val;
  }
  __shared__ float sh[256];
  __shared__ float sh2[256];
  sh[threadIdx.x] = s;
  sh2[threadIdx.x] = sq;
  __syncthreads();
  for (int d = 128; d > 0; d >>= 1) {
    if ((int)threadIdx.x < d) {
      sh[threadIdx.x] += sh[threadIdx.x + d];
      sh2[threadIdx.x] += sh2[threadIdx.x + d];
    }
    __syncthreads();
  }
  if (threadIdx.x == 0) {
    const float m = sh[0] / (float)PER;
    const float var = sh2[0] / (float)PER - m * m;
    mean[o] = m;
    rstd[o] = rsqrtf(var + EPSF);
  }
}

// ---------------------------------------------------------------------------
// Kernel 5: residual GEMM (res_w @ x[:,:,::2,:]) fused with BN + bias + ReLU.
// ---------------------------------------------------------------------------
__global__ __launch_bounds__(256) void k_final(const float* __restrict__ x,
                                               const float* __restrict__ rw,
                                               const float* __restrict__ rb,
                                               const float* __restrict__ xt,
                                               const float* __restrict__ mean,
                                               const float* __restrict__ rstd,
                                               const float* __restrict__ gamma,
                                               const float* __restrict__ beta,
                                               float* __restrict__ out) {
  const int L  = threadIdx.x & 31;
  const int wv = (blockIdx.x * blockDim.x + threadIdx.x) >> 5;
  const int colTile = wv % CT_TCN;
  const int rem     = wv / CT_TCN;
  const int oTile   = rem & 7;
  const int n       = rem >> 3;
  if (n >= Nb) return;

  const int o_a = oTile * 16 + (L & 15);
  const int col = colTile * 16 + (L & 15);
  const bool ok = col < COLS;
  const int tp  = col / Vv;
  const int v   = col % Vv;
  const int kh  = (L >> 4) * 2;
  const float* Bn = x + (size_t)n * Cc * TV;

  v8f acc = {};
  #pragma unroll 4
  for (int s = 0; s < Cc / 4; ++s) {
    const int k0 = s * 4 + kh;
    const v2f a = *(const v2f*)(rw + o_a * Cc + k0);
    v2f b;
    b.x = ok ? Bn[((size_t)(k0    ) * Tt + 2 * tp) * Vv + v] : 0.f;
    b.y = ok ? Bn[((size_t)(k0 + 1) * Tt + 2 * tp) * Vv + v] : 0.f;
    acc = WMMA4(a, b, acc);
  }

  if (ok) {
    #pragma unroll
    for (int r = 0; r < 8; ++r) {
      const int o = oTile * 16 + r + 8 * (L >> 4);
      const size_t oi = ((size_t)n * Oo + o) * COLS + col;
      const float bn = (xt[oi] - mean[o]) * rstd[o] * gamma[o] + beta[o];
      const float val = bn + acc[r] + rb[o];
      out[oi] = val > 0.f ? val : 0.f;
    }
  }
}

// ---------------------------------------------------------------------------
extern "C" void kernel_launch(void* const* d_in, const int* in_sizes, int n_in,
                              void* d_out, int out_size, void* d_ws, size_t ws_size,
                              hipStream_t stream) {
  const float* x     = (const float*)d_in[0];
  const float* Ap    = (const float*)d_in[1];
  const float* ei    = (const float*)d_in[2];
  const float* gw    = (const float*)d_in[3];
  const float* gb    = (const float*)d_in[4];
  const float* tw    = (const float*)d_in[5];
  const float* tb    = (const float*)d_in[6];
  const float* gamma = (const float*)d_in[7];
  const float* beta  = (const float*)d_in[8];
  const float* rw    = (const float*)d_in[9];
  const float* rb    = (const float*)d_in[10];
  float* out = (float*)d_out;

  float* ws   = (float*)d_ws;
  float* asum = ws + ASUM_OFF;
  float* mean = ws + MEAN_OFF;
  float* rstd = ws + RSTD_OFF;
  float* agg  = ws + AGG_OFF;
  float* xg   = ws + XGCN_OFF;
  float* twp  = ws + TWP_OFF;
  float* xt   = agg;  // agg is dead after k_gcn; reuse its region for x_tcn

  // 1 wave = one 16x16 output tile; 256 threads = 8 waves per block.
  const int aggWaves = (Nb * Cc * Tt / 16) * 2;       // 76800
  const int gcnWaves = Nb * 8 * CT_GCN;               // 120064
  const int tcnWaves = Nb * 8 * CT_TCN;               // 60160

  k_repack<<<(Oo * KT + 255) / 256, 256, 0, stream>>>(tw, twp);
  k_asum<<<1, 256, 0, stream>>>(Ap, ei, asum);
  k_agg<<<aggWaves / 8, 256, 0, stream>>>(x, asum, agg);
  k_gcn<<<(gcnWaves + 7) / 8, 256, 0, stream>>>(agg, gw, gb, xg);
  k_tcn<<<(tcnWaves + 7) / 8, 256, 0, stream>>>(xg, twp, tb, xt);
  k_bnstats<<<Oo, 256, 0, stream>>>(xt, mean, rstd);
  k_final<<<(tcnWaves + 7) / 8, 256, 0, stream>>>(x, rw, rb, xt, mean, rstd, gamma, beta, out);
}